// GAT_LSTM_88923002896915
// MI455X (gfx1250) — compile-verified
//
#include <hip/hip_runtime.h>
#include <math.h>

#define N_NODES 4096
#define N_EDGES 65536
#define BT      48          // B*T = 4*12
#define HG      64
#define NHEAD   8
#define KDIM    (N_NODES*HG)   // 262144
#define HL      128
#define G4      (4*HL)         // 512
#define TSTEPS  12
#define BATCH   4
#define OUTCOLS (N_NODES*8)    // 32768
#define KSPLIT  64

typedef __attribute__((ext_vector_type(2))) float v2f;
typedef __attribute__((ext_vector_type(8))) float v8f;

// ---------------- utility zero kernels ----------------
__global__ void zero_f32(float* p, int n) {
    int i = blockIdx.x * blockDim.x + threadIdx.x;
    if (i < n) p[i] = 0.f;
}
__global__ void zero_i32(int* p, int n) {
    int i = blockIdx.x * blockDim.x + threadIdx.x;
    if (i < n) p[i] = 0;
}

// ---------------- CSR build (edge list shared by all 48 graphs) ----------------
__global__ void edge_count(const int* __restrict__ dst, int* __restrict__ counts) {
    int e = blockIdx.x * blockDim.x + threadIdx.x;
    if (e < N_EDGES) atomicAdd(&counts[dst[e]], 1);
}

__global__ __launch_bounds__(1024) void scan4096(const int* __restrict__ counts,
                                                 int* __restrict__ row_off,
                                                 int* __restrict__ cursor) {
    __shared__ int lds[1024];
    int t = threadIdx.x;
    int l0 = counts[t*4+0], l1 = counts[t*4+1], l2 = counts[t*4+2], l3 = counts[t*4+3];
    int tot = l0 + l1 + l2 + l3;
    lds[t] = tot;
    __syncthreads();
    for (int off = 1; off < 1024; off <<= 1) {
        int v = (t >= off) ? lds[t - off] : 0;
        __syncthreads();
        lds[t] += v;
        __syncthreads();
    }
    int r = lds[t] - tot;   // exclusive prefix
    int base = t * 4;
    row_off[base+0] = r; cursor[base+0] = r; r += l0;
    row_off[base+1] = r; cursor[base+1] = r; r += l1;
    row_off[base+2] = r; cursor[base+2] = r; r += l2;
    row_off[base+3] = r; cursor[base+3] = r; r += l3;
    if (t == 1023) row_off[4096] = r;
}

__global__ void edge_scatter(const int* __restrict__ src, const int* __restrict__ dst,
                             int* __restrict__ cursor, int* __restrict__ csr_src) {
    int e = blockIdx.x * blockDim.x + threadIdx.x;
    if (e < N_EDGES) {
        int d = dst[e];
        int p = atomicAdd(&cursor[d], 1);
        csr_src[p] = src[e];
    }
}

// ---------------- GAT projection: h = x @ W_gat, plus attention dots ----------------
__global__ __launch_bounds__(256) void gat_project(const float* __restrict__ x,
                                                   const float* __restrict__ W_gat,
                                                   const float* __restrict__ att_src,
                                                   const float* __restrict__ att_dst,
                                                   float* __restrict__ h_feat,
                                                   float* __restrict__ a_s,
                                                   float* __restrict__ a_d) {
    __shared__ float Wg[8*64];
    __shared__ float As[64];
    __shared__ float Ad[64];
    int tid = threadIdx.x;
    for (int i = tid; i < 512; i += 256) Wg[i] = W_gat[i];
    if (tid < 64) { As[tid] = att_src[tid]; Ad[tid] = att_dst[tid]; }
    __syncthreads();

    int row = blockIdx.x * blockDim.x + tid;  // bt*N + n
    if (row >= BT * N_NODES) return;

    float xv[8];
#pragma unroll
    for (int f = 0; f < 8; ++f) xv[f] = x[(size_t)row * 8 + f];

    float hc[64];
#pragma unroll
    for (int c = 0; c < 64; ++c) {
        float s = 0.f;
#pragma unroll
        for (int f = 0; f < 8; ++f) s = fmaf(xv[f], Wg[f*64 + c], s);
        hc[c] = s;
        h_feat[(size_t)row * 64 + c] = s;
    }
#pragma unroll
    for (int hh = 0; hh < 8; ++hh) {
        float s1 = 0.f, s2 = 0.f;
#pragma unroll
        for (int c = 0; c < 8; ++c) {
            float v = hc[hh*8 + c];
            s1 = fmaf(v, As[hh*8 + c], s1);
            s2 = fmaf(v, Ad[hh*8 + c], s2);
        }
        a_s[(size_t)row * 8 + hh] = s1;
        a_d[(size_t)row * 8 + hh] = s2;
    }
}

// ---------------- GAT aggregation: one-pass online softmax per (bt, n, head) ----------------
__global__ __launch_bounds__(256) void gat_aggregate(const float* __restrict__ h_feat,
                                                     const float* __restrict__ a_s,
                                                     const float* __restrict__ a_d,
                                                     const int* __restrict__ row_off,
                                                     const int* __restrict__ csr_src,
                                                     const float* __restrict__ b_gat,
                                                     float* __restrict__ seq) {
    int idx = blockIdx.x * blockDim.x + threadIdx.x;   // < BT*N_NODES*8
    int hh = idx & 7;
    int n  = (idx >> 3) & (N_NODES - 1);
    int bt = idx >> 15;
    if (bt >= BT) return;

    size_t btBase   = (size_t)bt * N_NODES;
    size_t nodeBase = btBase + n;
    float adv = a_d[nodeBase * 8 + hh];

    // self-loop seeds the online softmax (guarantees denom > 0)
    float e0 = a_s[nodeBase * 8 + hh] + adv;
    e0 = e0 > 0.f ? e0 : 0.2f * e0;
    float m = e0, ssum = 1.0f;
    float acc[8];
    const float* hp = h_feat + nodeBase * 64 + (size_t)hh * 8;
#pragma unroll
    for (int c = 0; c < 8; ++c) acc[c] = hp[c];

    int beg = row_off[n], end = row_off[n + 1];
    for (int j = beg; j < end; ++j) {
        int s = csr_src[j];
        float e = a_s[(btBase + s) * 8 + hh] + adv;
        e = e > 0.f ? e : 0.2f * e;
        float p;
        if (e > m) {
            float sc = __expf(m - e);
            ssum *= sc;
#pragma unroll
            for (int c = 0; c < 8; ++c) acc[c] *= sc;
            m = e;
            p = 1.0f;
        } else {
            p = __expf(e - m);
        }
        ssum += p;
        const float* hs = h_feat + (btBase + s) * 64 + (size_t)hh * 8;
#pragma unroll
        for (int c = 0; c < 8; ++c) acc[c] = fmaf(p, hs[c], acc[c]);
    }

    float inv = 1.0f / ssum;
    float* so = seq + (size_t)bt * KDIM + (size_t)n * 64 + (size_t)hh * 8;
#pragma unroll
    for (int c = 0; c < 8; ++c) {
        float v = acc[c] * inv + b_gat[hh*8 + c];
        so[c] = v > 0.f ? v : (__expf(v) - 1.0f);     // ELU
    }
}

// ---------------- Gin[48,512] = seq @ W_ih^T  via v_wmma_f32_16x16x4_f32 ----------------
// A-tile = 16 rows of W_ih (M=j), B-tile = 16 rows of seq (N=bt), K streamed 4 at a time.
// ISA layout: lane m holds (K0,K1), lane m+16 holds (K2,K3) -> per-lane b64 loads at +0/+8B.
// Pointer-bump + 4x unroll so loads use immediate offsets instead of per-iter 64-bit math.
__global__ __launch_bounds__(128) void gemm_wih(const float* __restrict__ W_ih,
                                                const float* __restrict__ seq,
                                                float* __restrict__ Gin) {
    int jt   = blockIdx.x & 31;        // 32 j-tiles of 16 rows
    int kc   = blockIdx.x >> 5;        // KSPLIT k-chunks
    int wave = threadIdx.x >> 5;       // 4 waves split the chunk
    int lane = threadIdx.x & 31;
    int row  = lane & 15;              // M (and N) sub-index
    int hi   = lane >> 4;              // 0: K0,K1   1: K2,K3
    int j0   = jt * 16;

    const size_t kPerBlock = KDIM / KSPLIT;      // 4096
    const size_t kPerWave  = kPerBlock / 4;      // 1024
    size_t kbeg = (size_t)kc * kPerBlock + (size_t)wave * kPerWave;

    const float* Ap = W_ih + (size_t)(j0 + row) * KDIM + 2 * hi + kbeg;
    const float* B0 = seq + (size_t)row * KDIM + 2 * hi + kbeg;
    const float* B1 = B0 + (size_t)16 * KDIM;
    const float* B2 = B0 + (size_t)32 * KDIM;

    v8f c0 = {0,0,0,0,0,0,0,0};
    v8f c1 = {0,0,0,0,0,0,0,0};
    v8f c2 = {0,0,0,0,0,0,0,0};

    const int iters = (int)(kPerWave / 16);      // 64 outer iterations, 16 k each
    for (int it = 0; it < iters; ++it) {
        __builtin_prefetch(Ap + 1024, 0, 1);     // global_prefetch 4KB ahead on W_ih stream
#pragma unroll
        for (int u = 0; u < 4; ++u) {
            v2f a  = *(const v2f*)(Ap + u * 4);
            v2f b0 = *(const v2f*)(B0 + u * 4);
            v2f b1 = *(const v2f*)(B1 + u * 4);
            v2f b2 = *(const v2f*)(B2 + u * 4);
            c0 = __builtin_amdgcn_wmma_f32_16x16x4_f32(false, a, false, b0, (short)0, c0, false, false);
            c1 = __builtin_amdgcn_wmma_f32_16x16x4_f32(false, a, false, b1, (short)0, c1, false, false);
            c2 = __builtin_amdgcn_wmma_f32_16x16x4_f32(false, a, false, b2, (short)0, c2, false, false);
        }
        Ap += 16; B0 += 16; B1 += 16; B2 += 16;
    }

    // reduce 4 waves through LDS, then one set of global f32 atomics
    __shared__ float red[3 * 256];
    for (int i = threadIdx.x; i < 768; i += 128) red[i] = 0.f;
    __syncthreads();
#pragma unroll
    for (int r = 0; r < 8; ++r) {
        int jl = r + 8 * hi;          // D layout: lanes16-31 carry M = r+8
        atomicAdd(&red[0*256 + jl*16 + row], c0[r]);
        atomicAdd(&red[1*256 + jl*16 + row], c1[r]);
        atomicAdd(&red[2*256 + jl*16 + row], c2[r]);
    }
    __syncthreads();
    for (int i = threadIdx.x; i < 768; i += 128) {
        int t  = i >> 8;
        int rr = i & 255;
        int jl = rr >> 4;
        int cc = rr & 15;
        atomicAdd(&Gin[(size_t)(t*16 + cc) * G4 + j0 + jl], red[i]);
    }
}

// ---------------- sequential LSTM (B=4, Hl=128, T=12) in one block ----------------
__global__ __launch_bounds__(512) void lstm_kernel(const float* __restrict__ Gin,
                                                   const float* __restrict__ W_hh,
                                                   const float* __restrict__ b_ih,
                                                   const float* __restrict__ b_hh,
                                                   float* __restrict__ h_out) {
    __shared__ float hS[BATCH][HL];
    __shared__ float cS[BATCH][HL];
    __shared__ float gS[BATCH][G4];
    int j = threadIdx.x;  // 0..511
    if (j < HL) {
        for (int b = 0; b < BATCH; ++b) { hS[b][j] = 0.f; cS[b][j] = 0.f; }
    }
    __syncthreads();
    float bb = b_ih[j] + b_hh[j];
    for (int t = 0; t < TSTEPS; ++t) {
        float a0 = Gin[(size_t)(0*TSTEPS + t) * G4 + j] + bb;
        float a1 = Gin[(size_t)(1*TSTEPS + t) * G4 + j] + bb;
        float a2 = Gin[(size_t)(2*TSTEPS + t) * G4 + j] + bb;
        float a3 = Gin[(size_t)(3*TSTEPS + t) * G4 + j] + bb;
        const float* wr = W_hh + (size_t)j * HL;
        for (int k = 0; k < HL; ++k) {
            float w = wr[k];
            a0 = fmaf(hS[0][k], w, a0);
            a1 = fmaf(hS[1][k], w, a1);
            a2 = fmaf(hS[2][k], w, a2);
            a3 = fmaf(hS[3][k], w, a3);
        }
        gS[0][j] = a0; gS[1][j] = a1; gS[2][j] = a2; gS[3][j] = a3;
        __syncthreads();
        if (j < HL) {
            for (int b = 0; b < BATCH; ++b) {
                float ig = 1.f / (1.f + __expf(-gS[b][j]));
                float fg = 1.f / (1.f + __expf(-gS[b][HL + j]));
                float gg = tanhf(gS[b][2*HL + j]);
                float og = 1.f / (1.f + __expf(-gS[b][3*HL + j]));
                float c  = fg * cS[b][j] + ig * gg;
                cS[b][j] = c;
                hS[b][j] = og * tanhf(c);
            }
        }
        __syncthreads();
    }
    if (j < HL)
        for (int b = 0; b < BATCH; ++b) h_out[b*HL + j] = hS[b][j];
}

// ---------------- final FC: out[b, 32768] = h_last @ W_fc + b_fc ----------------
__global__ __launch_bounds__(256) void fc_kernel(const float* __restrict__ hlast,
                                                 const float* __restrict__ W_fc,
                                                 const float* __restrict__ b_fc,
                                                 float* __restrict__ out) {
    __shared__ float hl[BATCH * HL];
    for (int i = threadIdx.x; i < BATCH * HL; i += blockDim.x) hl[i] = hlast[i];
    __syncthreads();
    int col = blockIdx.x * blockDim.x + threadIdx.x;
    if (col >= OUTCOLS) return;
    float a0 = b_fc[col], a1 = a0, a2 = a0, a3 = a0;
    for (int k = 0; k < HL; ++k) {
        float w = W_fc[(size_t)k * OUTCOLS + col];
        a0 = fmaf(hl[0*HL + k], w, a0);
        a1 = fmaf(hl[1*HL + k], w, a1);
        a2 = fmaf(hl[2*HL + k], w, a2);
        a3 = fmaf(hl[3*HL + k], w, a3);
    }
    out[(size_t)0*OUTCOLS + col] = a0;
    out[(size_t)1*OUTCOLS + col] = a1;
    out[(size_t)2*OUTCOLS + col] = a2;
    out[(size_t)3*OUTCOLS + col] = a3;
}

extern "C" void kernel_launch(void* const* d_in, const int* in_sizes, int n_in,
                              void* d_out, int out_size, void* d_ws, size_t ws_size,
                              hipStream_t stream) {
    const float* x       = (const float*)d_in[0];
    const int*   edge    = (const int*)d_in[1];     // [2, 65536]: row0=src, row1=dst
    const float* W_gat   = (const float*)d_in[2];
    const float* att_src = (const float*)d_in[3];
    const float* att_dst = (const float*)d_in[4];
    const float* b_gat   = (const float*)d_in[5];
    const float* W_ih    = (const float*)d_in[6];
    const float* W_hh    = (const float*)d_in[7];
    const float* b_ih    = (const float*)d_in[8];
    const float* b_hh    = (const float*)d_in[9];
    const float* W_fc    = (const float*)d_in[10];
    const float* b_fc    = (const float*)d_in[11];
    float* out = (float*)d_out;

    // workspace carve-up
    float* wsf   = (float*)d_ws;
    size_t off   = 0;
    float* h_feat = wsf + off; off += (size_t)BT * N_NODES * HG;   // 12.58M
    float* a_s    = wsf + off; off += (size_t)BT * N_NODES * 8;    // 1.57M
    float* a_d    = wsf + off; off += (size_t)BT * N_NODES * 8;    // 1.57M
    float* seq    = wsf + off; off += (size_t)BT * KDIM;           // 12.58M
    float* Gin    = wsf + off; off += (size_t)BT * G4;             // 24576
    float* hlast  = wsf + off; off += 512;
    int* counts  = (int*)(wsf + off);
    int* row_off = counts + 4096;
    int* cursor  = row_off + 4104;   // 4097 used, padded
    int* csr_src = cursor + 4096;

    const int* e_src = edge;
    const int* e_dst = edge + N_EDGES;

    zero_i32<<<(4096 + 255) / 256, 256, 0, stream>>>(counts, 4096);
    edge_count<<<N_EDGES / 256, 256, 0, stream>>>(e_dst, counts);
    scan4096<<<1, 1024, 0, stream>>>(counts, row_off, cursor);
    edge_scatter<<<N_EDGES / 256, 256, 0, stream>>>(e_src, e_dst, cursor, csr_src);

    gat_project<<<(BT * N_NODES) / 256, 256, 0, stream>>>(x, W_gat, att_src, att_dst,
                                                          h_feat, a_s, a_d);
    gat_aggregate<<<(BT * N_NODES * 8) / 256, 256, 0, stream>>>(h_feat, a_s, a_d,
                                                                row_off, csr_src, b_gat, seq);

    zero_f32<<<(BT * G4 + 255) / 256, 256, 0, stream>>>(Gin, BT * G4);
    gemm_wih<<<32 * KSPLIT, 128, 0, stream>>>(W_ih, seq, Gin);

    lstm_kernel<<<1, 512, 0, stream>>>(Gin, W_hh, b_ih, b_hh, hlast);
    fc_kernel<<<(OUTCOLS + 255) / 256, 256, 0, stream>>>(hlast, W_fc, b_fc, out);
}